// MultiModalFusionGAT_2473901163079
// MI455X (gfx1250) — compile-verified
//
#include <hip/hip_runtime.h>
#include <hip/hip_bf16.h>
#include <math.h>

typedef __attribute__((ext_vector_type(2))) float v2f;
typedef __attribute__((ext_vector_type(8))) float v8f;
typedef int vsi4 __attribute__((vector_size(4 * sizeof(int))));

#define HID   256
#define HEADS 8
#define N_USER 10000
#define N_ITEM 30000
#define N_MOD  30000
#define N_DST  (N_USER + N_ITEM)            // 40000 (only user/item receive edges)
#define N_TOT  (N_USER + N_ITEM + 3*N_MOD)  // 130000

#define BK        64     // K panel
#define SA_STRIDE 68     // padded A row stride (words): 16B-aligned, bank-safe
#define SW_STRIDE 80     // padded W row stride (words): 16B-aligned, half-waves disjoint banks

// ---------------------------------------------------------------------------
// CDNA5 async global->LDS copy (ASYNCcnt-tracked) + wait
// ---------------------------------------------------------------------------
__device__ __forceinline__ void async_copy_b128(const float* g, float* l) {
#if __has_builtin(__builtin_amdgcn_global_load_async_to_lds_b128)
    __builtin_amdgcn_global_load_async_to_lds_b128((vsi4*)g, (vsi4*)l, 0, 0);
#else
    asm volatile("global_load_async_to_lds_b128 %0, %1, off"
                 :: "v"((unsigned)(unsigned long long)l), "v"(g) : "memory");
#endif
}
__device__ __forceinline__ void wait_async0() {
#if __has_builtin(__builtin_amdgcn_s_wait_asynccnt)
    __builtin_amdgcn_s_wait_asynccnt(0);
#else
    asm volatile("s_wait_asynccnt 0x0" ::: "memory");
#endif
}

// ---------------------------------------------------------------------------
// Utility: float atomic max via signed-max / unsigned-min trick (init = -inf)
// ---------------------------------------------------------------------------
__device__ __forceinline__ void atomicMaxF(float* addr, float val) {
    if (val >= 0.0f) atomicMax((int*)addr, __float_as_int(val));
    else             atomicMin((unsigned int*)addr, (unsigned int)__float_as_int(val));
}

// ---------------------------------------------------------------------------
// Init: aggr = 0, asum = 0, amax = -inf (must run every call: graph replay)
// ---------------------------------------------------------------------------
__global__ __launch_bounds__(256) void init_state_kernel(float* aggr, float* asum, float* amax) {
    int idx = blockIdx.x * blockDim.x + threadIdx.x;
    if (idx < N_DST * HID) aggr[idx] = 0.0f;
    if (idx < N_DST * HEADS) {
        asum[idx] = 0.0f;
        amax[idx] = __int_as_float(0xFF800000); // -inf
    }
}

// ---------------------------------------------------------------------------
// Fuse Wkqv (k,v slices) with per-head relation matrices:
//   Wkp[e][r][h*32+j] = sum_d Wkqv[s][r][h*32+d]     * Wk_rel[e][h][d][j]
//   Wvp[e][r][h*32+j] = sum_d Wkqv[s][r][512+h*32+d] * Wv_rel[e][h][d][j]
// row r == 256 computes the fused biases from bkqv.
// ---------------------------------------------------------------------------
__global__ __launch_bounds__(256) void fuse_rel_weights_kernel(
    const float* __restrict__ Wkqv, const float* __restrict__ bkqv,
    const float* __restrict__ Wk_rel, const float* __restrict__ Wv_rel,
    float* __restrict__ Wkp, float* __restrict__ bkp,
    float* __restrict__ Wvp, float* __restrict__ bvp)
{
    int idx = blockIdx.x * blockDim.x + threadIdx.x;
    const int total = 5 * 257 * 256;
    if (idx >= total) return;
    int c = idx & 255;
    int t = idx >> 8;
    int r = t % 257;
    int e = t / 257;
    const int SRC[5] = {2, 3, 4, 0, 1};
    int s = SRC[e];
    int h = c >> 5, j = c & 31;
    const float* wk = Wk_rel + (size_t)((e * 8 + h) * 32) * 32 + j; // stride 32 over d
    const float* wv = Wv_rel + (size_t)((e * 8 + h) * 32) * 32 + j;
    float aK = 0.0f, aV = 0.0f;
    if (r < 256) {
        const float* rowp = Wkqv + (size_t)(s * 256 + r) * 768;
        #pragma unroll 8
        for (int d = 0; d < 32; ++d) {
            aK += rowp[h * 32 + d]       * wk[d * 32];
            aV += rowp[512 + h * 32 + d] * wv[d * 32];
        }
        Wkp[((size_t)e << 16) + (r << 8) + c] = aK;
        Wvp[((size_t)e << 16) + (r << 8) + c] = aV;
    } else {
        const float* bp = bkqv + (size_t)s * 768;
        #pragma unroll 8
        for (int d = 0; d < 32; ++d) {
            aK += bp[h * 32 + d]       * wk[d * 32];
            aV += bp[512 + h * 32 + d] * wv[d * 32];
        }
        bkp[e * 256 + c] = aK;
        bvp[e * 256 + c] = aV;
    }
}

// ---------------------------------------------------------------------------
// WMMA f32 GEMM core, LDS-staged via async-to-LDS.
// C[M x 256] = A[M x 256(lda)] @ W[256 x 256(ldw)]   (bias/epilogue by caller)
// Block = 256 threads (8 waves). Wave: 16 rows x 64 cols (4 C accumulators).
// grid = (ceil(M/128), 4). K consumed in panels of 64 from LDS.
// ---------------------------------------------------------------------------
__device__ __forceinline__ void gemm_core(
    const float* __restrict__ A, int lda,
    const float* __restrict__ W, int ldw,
    int M, v8f acc[4])
{
    __shared__ float sW[BK * SW_STRIDE];            // 20 KB: W panel [kk][col]
    __shared__ float sA[8 * 16 * SA_STRIDE];        // 34 KB: per-wave A panels [r][kk]

    int wave = threadIdx.x >> 5;
    int lane = threadIdx.x & 31;
    int rowbase = blockIdx.x * 128 + wave * 16;
    int colbase = blockIdx.y * 64;
    int l16 = lane & 15;
    int khi = (lane >> 4) << 1;                     // 0 (lanes 0-15) or 2 (lanes 16-31)

    for (int kp = 0; kp < HID; kp += BK) {
        // Stage W panel: BK rows x 64 cols = 1024 float4 chunks across 256 threads
        for (int i = threadIdx.x; i < BK * 16; i += 256) {
            int kk = i >> 4, cg = i & 15;
            async_copy_b128(W + (size_t)(kp + kk) * ldw + colbase + cg * 4,
                            &sW[kk * SW_STRIDE + cg * 4]);
        }
        // Stage this wave's A panel: 16 rows x BK = 256 float4 chunks across 32 lanes
        for (int i = lane; i < 16 * (BK / 4); i += 32) {
            int r = i >> 4, cg = i & 15;
            int row = rowbase + r; if (row >= M) row = M - 1;
            async_copy_b128(A + (size_t)row * lda + kp + cg * 4,
                            &sA[(wave * 16 + r) * SA_STRIDE + cg * 4]);
        }
        wait_async0();
        __syncthreads();

        #pragma unroll
        for (int k0 = 0; k0 < BK; k0 += 4) {
            v2f a = *(const v2f*)&sA[(wave * 16 + l16) * SA_STRIDE + k0 + khi];
            const float* Wl = &sW[(k0 + khi) * SW_STRIDE + l16];
            v2f b0; b0.x = Wl[0];  b0.y = Wl[SW_STRIDE + 0];
            v2f b1; b1.x = Wl[16]; b1.y = Wl[SW_STRIDE + 16];
            v2f b2; b2.x = Wl[32]; b2.y = Wl[SW_STRIDE + 32];
            v2f b3; b3.x = Wl[48]; b3.y = Wl[SW_STRIDE + 48];
            acc[0] = __builtin_amdgcn_wmma_f32_16x16x4_f32(false, a, false, b0, (short)0, acc[0], false, false);
            acc[1] = __builtin_amdgcn_wmma_f32_16x16x4_f32(false, a, false, b1, (short)0, acc[1], false, false);
            acc[2] = __builtin_amdgcn_wmma_f32_16x16x4_f32(false, a, false, b2, (short)0, acc[2], false, false);
            acc[3] = __builtin_amdgcn_wmma_f32_16x16x4_f32(false, a, false, b3, (short)0, acc[3], false, false);
        }
        __syncthreads();
    }
}

// C = A @ W + bias
__global__ __launch_bounds__(256) void gemm_bias_kernel(
    const float* __restrict__ A, int lda,
    const float* __restrict__ W, int ldw,
    const float* __restrict__ bias,
    float* __restrict__ C, int ldc, int M)
{
    v8f acc[4] = {};
    gemm_core(A, lda, W, ldw, M, acc);

    int wave = threadIdx.x >> 5;
    int lane = threadIdx.x & 31;
    int l16 = lane & 15;
    int rowbase = blockIdx.x * 128 + wave * 16;
    int colbase = blockIdx.y * 64;
    int rbase = rowbase + ((lane >> 4) << 3);       // C layout: VGPR r -> M = r + (lane<16?0:8)
    #pragma unroll
    for (int nt = 0; nt < 4; ++nt) {
        int col = colbase + nt * 16 + l16;
        float bv = bias[col];
        #pragma unroll
        for (int r = 0; r < 8; ++r) {
            int row = rbase + r;
            if (row < M) C[(size_t)row * ldc + col] = acc[nt][r] + bv;
        }
    }
}

// out = sg * (A @ Wout + bout) + (1-sg) * xres,  sg = sigmoid(skip[j])
__global__ __launch_bounds__(256) void gemm_out_kernel(
    const float* __restrict__ A,
    const float* __restrict__ W,
    const float* __restrict__ bias,
    const float* __restrict__ xres,
    const float* __restrict__ skipPtr,
    float* __restrict__ out, int M)
{
    v8f acc[4] = {};
    gemm_core(A, HID, W, HID, M, acc);

    float sg = 1.0f / (1.0f + __expf(-skipPtr[0]));
    float og = 1.0f - sg;

    int wave = threadIdx.x >> 5;
    int lane = threadIdx.x & 31;
    int l16 = lane & 15;
    int rowbase = blockIdx.x * 128 + wave * 16;
    int colbase = blockIdx.y * 64;
    int rbase = rowbase + ((lane >> 4) << 3);
    #pragma unroll
    for (int nt = 0; nt < 4; ++nt) {
        int col = colbase + nt * 16 + l16;
        float bv = bias[col];
        #pragma unroll
        for (int r = 0; r < 8; ++r) {
            int row = rbase + r;
            if (row < M) {
                float o = acc[nt][r] + bv;
                out[(size_t)row * HID + col] = sg * o + og * xres[(size_t)row * HID + col];
            }
        }
    }
}

// ---------------------------------------------------------------------------
// Edge pass 1: logits a[e,h] = dot(q[dst,h], kp[src,h]) * p_rel[h] * scale
// One wave per edge; lane covers 8 contiguous floats (head = lane>>2).
// ---------------------------------------------------------------------------
__global__ __launch_bounds__(256) void edge_logits_kernel(
    const int* __restrict__ ei, int E,
    const float* __restrict__ qbuf, const float* __restrict__ kpbuf,
    const float* __restrict__ prel, float scale,
    int soff, int doff, float* __restrict__ alpha, long long abase,
    float* __restrict__ amax)
{
    int widx = (blockIdx.x * blockDim.x + threadIdx.x) >> 5;
    int lane = threadIdx.x & 31;
    if (widx >= E) return;
    int src = ei[widx];
    int dst = ei[E + widx];
    const float4* qv = (const float4*)(qbuf + ((size_t)(dst + doff)) * HID + lane * 8);
    const float4* kv = (const float4*)(kpbuf + ((size_t)(src + soff)) * HID + lane * 8);
    float4 q0 = qv[0], q1 = qv[1];
    float4 k0 = kv[0], k1 = kv[1];
    float s = q0.x * k0.x + q0.y * k0.y + q0.z * k0.z + q0.w * k0.w
            + q1.x * k1.x + q1.y * k1.y + q1.z * k1.z + q1.w * k1.w;
    s += __shfl_xor(s, 1);
    s += __shfl_xor(s, 2);                       // lanes 4h..4h+3 share head-h dot
    int head = lane >> 2;
    float a = s * prel[head] * scale;
    if ((lane & 3) == 0) {
        alpha[(abase + widx) * HEADS + head] = a;
        atomicMaxF(&amax[(size_t)(dst + doff) * HEADS + head], a);
    }
}

// ---------------------------------------------------------------------------
// Edge pass 2: ae = exp(a - amax[dst,h]); asum[dst,h] += ae;
//              aggr[dst,:] += ae * vp[src,:]   (normalization deferred)
// ---------------------------------------------------------------------------
__global__ __launch_bounds__(256) void edge_aggr_kernel(
    const int* __restrict__ ei, int E,
    const float* __restrict__ vpbuf,
    const float* __restrict__ alpha, long long abase,
    const float* __restrict__ amax, float* __restrict__ asum,
    int soff, int doff, float* __restrict__ aggr)
{
    int widx = (blockIdx.x * blockDim.x + threadIdx.x) >> 5;
    int lane = threadIdx.x & 31;
    if (widx >= E) return;
    int src = ei[widx];
    int dst = ei[E + widx];
    int head = lane >> 2;
    float a = alpha[(abase + widx) * HEADS + head];
    float m = amax[(size_t)(dst + doff) * HEADS + head];
    float ae = __expf(a - m);
    if ((lane & 3) == 0)
        atomicAdd(&asum[(size_t)(dst + doff) * HEADS + head], ae);
    const float4* vv = (const float4*)(vpbuf + ((size_t)(src + soff)) * HID + lane * 8);
    float* og = aggr + ((size_t)(dst + doff)) * HID + lane * 8;
    float4 v0 = vv[0], v1 = vv[1];
    atomicAdd(og + 0, ae * v0.x);
    atomicAdd(og + 1, ae * v0.y);
    atomicAdd(og + 2, ae * v0.z);
    atomicAdd(og + 3, ae * v0.w);
    atomicAdd(og + 4, ae * v1.x);
    atomicAdd(og + 5, ae * v1.y);
    atomicAdd(og + 6, ae * v1.z);
    atomicAdd(og + 7, ae * v1.w);
}

// ---------------------------------------------------------------------------
// Normalize by asum and apply exact GELU -> g buffer (GEMM input)
// ---------------------------------------------------------------------------
__global__ __launch_bounds__(256) void norm_gelu_kernel(
    const float* __restrict__ aggr, const float* __restrict__ asum,
    float* __restrict__ g)
{
    int idx = blockIdx.x * blockDim.x + threadIdx.x;
    if (idx >= N_DST * HID) return;
    int n = idx >> 8;
    int c = idx & 255;
    int h = c >> 5;
    float s = asum[(size_t)n * HEADS + h] + 1e-16f;
    float v = aggr[idx] / s;
    float ge = 0.5f * v * (1.0f + erff(v * 0.70710678118654752f));
    g[idx] = ge;
}

// ---------------------------------------------------------------------------
// Host-side launcher
// ---------------------------------------------------------------------------
extern "C" void kernel_launch(void* const* d_in, const int* in_sizes, int n_in,
                              void* d_out, int out_size, void* d_ws, size_t ws_size,
                              hipStream_t stream) {
    const float* x_user  = (const float*)d_in[0];
    const float* x_item  = (const float*)d_in[1];
    const float* x_taste = (const float*)d_in[2];
    const float* x_inten = (const float*)d_in[3];
    const float* x_image = (const float*)d_in[4];
    const float* Wkqv    = (const float*)d_in[5];
    const float* bkqv    = (const float*)d_in[6];
    const float* Wk_rel  = (const float*)d_in[7];
    const float* Wv_rel  = (const float*)d_in[8];
    const float* p_rel   = (const float*)d_in[9];
    const float* Wout    = (const float*)d_in[10];
    const float* bout    = (const float*)d_in[11];
    const float* skip    = (const float*)d_in[12];
    const int* ei[5] = {(const int*)d_in[13], (const int*)d_in[14], (const int*)d_in[15],
                        (const int*)d_in[16], (const int*)d_in[17]};
    int E[5];
    for (int e = 0; e < 5; ++e) E[e] = in_sizes[13 + e] / 2;

    // Workspace layout (floats)
    float* ws = (float*)d_ws;
    size_t off = 0;
    float* Wkp   = ws + off; off += 5 * 256 * 256;
    float* bkp   = ws + off; off += 5 * 256;
    float* Wvp   = ws + off; off += 5 * 256 * 256;
    float* bvp   = ws + off; off += 5 * 256;
    float* qbuf  = ws + off; off += (size_t)N_DST * HID;   // user rows 0..9999, item 10000..39999
    float* kpbuf = ws + off; off += (size_t)N_TOT * HID;   // global node order
    float* vpbuf = ws + off; off += (size_t)N_TOT * HID;
    long long Etot = 0; for (int e = 0; e < 5; ++e) Etot += E[e];
    float* alpha = ws + off; off += (size_t)Etot * HEADS;
    float* amax  = ws + off; off += (size_t)N_DST * HEADS;
    float* asum  = ws + off; off += (size_t)N_DST * HEADS;
    float* aggr  = ws + off; off += (size_t)N_DST * HID;
    float* gbuf  = ws + off; off += (size_t)N_DST * HID;
    if (ws_size < off * sizeof(float)) return;

    const float scale = 0.17677669529663688f;  // 1/sqrt(32)

    const float* xs[5] = {x_user, x_item, x_taste, x_inten, x_image};
    const int    Ms[5] = {N_USER, N_ITEM, N_MOD, N_MOD, N_MOD};
    const int  toff[5] = {0, N_USER, N_DST, N_DST + N_MOD, N_DST + 2 * N_MOD};
    const int  SRC[5]  = {2, 3, 4, 0, 1};
    const int  DOFF[5] = {N_USER, N_USER, N_USER, N_USER, 0};

    // 1) init accumulators
    {
        int total = N_DST * HID;
        init_state_kernel<<<(total + 255) / 256, 256, 0, stream>>>(aggr, asum, amax);
    }
    // 2) fuse relation weights into 256x256 projections
    {
        int total = 5 * 257 * 256;
        fuse_rel_weights_kernel<<<(total + 255) / 256, 256, 0, stream>>>(
            Wkqv, bkqv, Wk_rel, Wv_rel, Wkp, bkp, Wvp, bvp);
    }
    // 3) q projections for dst types 0,1 (slice of Wkqv, ldw=768)
    for (int i = 0; i < 2; ++i) {
        int M = Ms[i];
        dim3 grid((M + 127) / 128, 4);
        gemm_bias_kernel<<<grid, 256, 0, stream>>>(
            xs[i], HID,
            Wkqv + (size_t)i * 256 * 768 + 256, 768,
            bkqv + (size_t)i * 768 + 256,
            qbuf + (size_t)toff[i] * HID, HID, M);
    }
    // 4) fused kp / vp projections (one per edge type == one per source type)
    for (int e = 0; e < 5; ++e) {
        int s = SRC[e];
        int M = Ms[s];
        dim3 grid((M + 127) / 128, 4);
        gemm_bias_kernel<<<grid, 256, 0, stream>>>(
            xs[s], HID, Wkp + ((size_t)e << 16), 256, bkp + e * 256,
            kpbuf + (size_t)toff[s] * HID, HID, M);
        gemm_bias_kernel<<<grid, 256, 0, stream>>>(
            xs[s], HID, Wvp + ((size_t)e << 16), 256, bvp + e * 256,
            vpbuf + (size_t)toff[s] * HID, HID, M);
    }
    // 5) edge logits + segment max
    {
        long long abase = 0;
        for (int e = 0; e < 5; ++e) {
            int blocks = (E[e] + 7) / 8;  // 8 waves (edges) per 256-thread block
            edge_logits_kernel<<<blocks, 256, 0, stream>>>(
                ei[e], E[e], qbuf, kpbuf, p_rel + e * HEADS, scale,
                toff[SRC[e]], DOFF[e], alpha, abase, amax);
            abase += E[e];
        }
    }
    // 6) edge exp / sum / weighted aggregation
    {
        long long abase = 0;
        for (int e = 0; e < 5; ++e) {
            int blocks = (E[e] + 7) / 8;
            edge_aggr_kernel<<<blocks, 256, 0, stream>>>(
                ei[e], E[e], vpbuf, alpha, abase, amax, asum,
                toff[SRC[e]], DOFF[e], aggr);
            abase += E[e];
        }
    }
    // 7) normalize + GELU
    {
        int total = N_DST * HID;
        norm_gelu_kernel<<<(total + 255) / 256, 256, 0, stream>>>(aggr, asum, gbuf);
    }
    // 8) output GEMMs with sigmoid-skip blend -> d_out (res0 | res1)
    {
        float* outp = (float*)d_out;
        dim3 g0((N_USER + 127) / 128, 4);
        gemm_out_kernel<<<g0, 256, 0, stream>>>(
            gbuf, Wout, bout, x_user, skip, outp, N_USER);
        dim3 g1((N_ITEM + 127) / 128, 4);
        gemm_out_kernel<<<g1, 256, 0, stream>>>(
            gbuf + (size_t)N_USER * HID, Wout + 256 * 256, bout + 256,
            x_item, skip + 1, outp + (size_t)N_USER * HID, N_ITEM);
    }
}